// lstm_reg_14302241096311
// MI455X (gfx1250) — compile-verified
//
#include <hip/hip_runtime.h>
#include <hip/hip_bf16.h>

// Problem constants (match the JAX reference).
#define SEQ   512
#define BATCH 256
#define INF   180   // input features
#define HD    8     // hidden size
#define G4    32    // 4*H gates

typedef __attribute__((ext_vector_type(2))) float v2f;
typedef __attribute__((ext_vector_type(8))) float v8f;

// ---------------------------------------------------------------------------
// Kernel 1: xg0 = x @ w_ih0^T + (b_ih0 + b_hh0), f32 WMMA, 2x2 tile per wave.
//   x  : [S*B, 180] fp32 row-major
//   w  : [32, 180]  fp32 row-major (gate-major: i 0-7, f 8-15, g 16-23, o 24-31)
//   xg : [S*B][8][4] fp32 -- gate-interleaved: unit j holds (i,f,g,o) contiguous
//
// One wave computes a 32-row x 32-gate block of the [131072 x 32] result:
//   4 accumulators (acc00, acc01, acc10, acc11), K-loop 45 x (2 A-loads,
//   2 B-loads, 4 v_wmma_f32_16x16x4_f32). Each x element is loaded exactly
//   once device-wide -> DRAM traffic = 94 MB (the roofline floor, ~4 us).
//
// f32 WMMA fragment layouts (ISA 7.12.2):
//   A 16x4 : lane L holds (M=L%16, K=k0+2*(L/16)+{0,1})  -> contiguous float2
//   B 4x16 : lane L holds (K=k0+2*(L/16)+{0,1}, N=L%16)  -> contiguous float2
//   C 16x16: vgpr r, lane L -> (M = r + 8*(L/16), N = L%16)
// ---------------------------------------------------------------------------
__global__ __launch_bounds__(256) void xgemm_wmma_kernel(
    const float* __restrict__ x,
    const float* __restrict__ w_ih0,
    const float* __restrict__ b_ih0,
    const float* __restrict__ b_hh0,
    float* __restrict__ xg)
{
    const int gwave = (int)((blockIdx.x * blockDim.x + threadIdx.x) >> 5); // 0..4095
    const int lane  = (int)(threadIdx.x & 31);
    const int n     = lane & 15;
    const int hi    = lane >> 4;

    const int rowbase = gwave * 32;
    const float* a0p = x + (size_t)(rowbase + n)      * INF;  // A rows 0..15
    const float* a1p = x + (size_t)(rowbase + 16 + n) * INF;  // A rows 16..31
    const float* b0p = w_ih0 + (size_t)n        * INF;        // gates 0..15
    const float* b1p = w_ih0 + (size_t)(n + 16) * INF;        // gates 16..31

    v8f acc00 = {0.f,0.f,0.f,0.f,0.f,0.f,0.f,0.f};
    v8f acc01 = acc00, acc10 = acc00, acc11 = acc00;

    #pragma unroll 5
    for (int k0 = 0; k0 < INF; k0 += 4) {
        const int k = k0 + 2 * hi;
        const v2f a0 = *(const v2f*)(a0p + k);   // 8B aligned: row pitch 720B
        const v2f a1 = *(const v2f*)(a1p + k);
        const v2f b0 = *(const v2f*)(b0p + k);
        const v2f b1 = *(const v2f*)(b1p + k);
        acc00 = __builtin_amdgcn_wmma_f32_16x16x4_f32(false, a0, false, b0,
                                                      (short)0, acc00, false, false);
        acc01 = __builtin_amdgcn_wmma_f32_16x16x4_f32(false, a0, false, b1,
                                                      (short)0, acc01, false, false);
        acc10 = __builtin_amdgcn_wmma_f32_16x16x4_f32(false, a1, false, b0,
                                                      (short)0, acc10, false, false);
        acc11 = __builtin_amdgcn_wmma_f32_16x16x4_f32(false, a1, false, b1,
                                                      (short)0, acc11, false, false);
    }

    // Column index of this lane: gate n (acc*0) and gate n+16 (acc*1).
    const float bias0 = b_ih0[n]      + b_hh0[n];
    const float bias1 = b_ih0[n + 16] + b_hh0[n + 16];
    // Gate-interleaved position: p = (gate%8)*4 + gate/8 ; note p1 = p0 + 2.
    const int p0 = (n & 7) * 4 + (n >> 3);
    const int p1 = p0 + 2;

    float* ob = xg + (size_t)(rowbase + hi * 8) * G4;
    #pragma unroll
    for (int r = 0; r < 8; ++r) {
        ob[(size_t)r * G4 + p0]        = acc00[r] + bias0;   // rows hi*8+r
        ob[(size_t)r * G4 + p1]        = acc01[r] + bias1;
        ob[(size_t)(r + 16) * G4 + p0] = acc10[r] + bias0;   // rows 16+hi*8+r
        ob[(size_t)(r + 16) * G4 + p1] = acc11[r] + bias1;
    }
}

// ---------------------------------------------------------------------------
// Kernel 2: fused two-layer LSTM scan + regression head.
// 8 lanes per batch element (lane j = hidden unit j, owns its 4 gates in both
// layers). 2048 threads = 32 blocks x 64 (2 waves) for WGP spread. Hidden
// state broadcast is intra-wave ds_bpermute within aligned 8-lane groups --
// no barriers inside the 512-step loop. xg load for step s+1 is prefetched
// while step s computes.
// ---------------------------------------------------------------------------
__device__ __forceinline__ float fast_sigmoid(float x) {
    return 1.0f / (1.0f + __expf(-x));
}
__device__ __forceinline__ float fast_tanh(float x) {
    // 1 - 2/(e^{2x}+1); saturates correctly for |x| large.
    return 1.0f - 2.0f / (__expf(2.0f * x) + 1.0f);
}

__global__ __launch_bounds__(64) void lstm_scan_kernel(
    const float* __restrict__ xg,      // [S*B][32] gate-interleaved
    const float* __restrict__ w_hh0,   // [32][8]
    const float* __restrict__ w_ih1,   // [32][8]
    const float* __restrict__ w_hh1,   // [32][8]
    const float* __restrict__ b_ih1,   // [32]
    const float* __restrict__ b_hh1,   // [32]
    const float* __restrict__ w_reg,   // [2][8]
    const float* __restrict__ b_reg,   // [2]
    float* __restrict__ out)           // [S*B][2]
{
    const int tid  = (int)threadIdx.x;
    const int j    = tid & 7;                      // hidden unit
    const int b    = (int)blockIdx.x * 8 + (tid >> 3);
    const int base = tid & 24;                     // lane-group base within wave

    // Per-lane weight rows: gate g of unit j is row g*8+j of each [32][8] matrix.
    float w0[4][HD], wi1[4][HD], wh1[4][HD], bias1[4];
    #pragma unroll
    for (int g = 0; g < 4; ++g) {
        const int r = g * 8 + j;
        #pragma unroll
        for (int k = 0; k < HD; ++k) {
            w0 [g][k] = w_hh0[r * HD + k];
            wi1[g][k] = w_ih1[r * HD + k];
            wh1[g][k] = w_hh1[r * HD + k];
        }
        bias1[g] = b_ih1[r] + b_hh1[r];
    }
    const float wr0 = w_reg[j], wr1 = w_reg[HD + j];
    const float br0 = b_reg[0], br1 = b_reg[1];

    float h1 = 0.f, c1 = 0.f, h2 = 0.f, c2 = 0.f;

    const float4* xgp = (const float4*)(xg + (size_t)b * G4 + j * 4);
    const size_t xstride = (size_t)BATCH * G4 / 4;   // float4 elements per step

    float4 gin = xgp[0];                             // prefetched step 0
    for (int s = 0; s < SEQ; ++s) {
        // Prefetch next step (wraps to 0 on the last iteration; harmless).
        const float4 gnext = xgp[(size_t)((s + 1) & (SEQ - 1)) * xstride];

        // ----- layer 0 -----
        float gi = gin.x, gf = gin.y, gg = gin.z, go = gin.w;
        #pragma unroll
        for (int k = 0; k < HD; ++k) {
            const float hk = __shfl(h1, base + k, 32);
            gi += hk * w0[0][k];
            gf += hk * w0[1][k];
            gg += hk * w0[2][k];
            go += hk * w0[3][k];
        }
        const float i0 = fast_sigmoid(gi);
        const float f0 = fast_sigmoid(gf);
        const float g0 = fast_tanh(gg);
        const float o0 = fast_sigmoid(go);
        c1 = f0 * c1 + i0 * g0;
        h1 = o0 * fast_tanh(c1);

        // ----- layer 1 (input = h1 of this step) -----
        float a0 = bias1[0], a1 = bias1[1], a2 = bias1[2], a3 = bias1[3];
        #pragma unroll
        for (int k = 0; k < HD; ++k) {
            const float h1k = __shfl(h1, base + k, 32);
            const float h2k = __shfl(h2, base + k, 32);
            a0 += h1k * wi1[0][k] + h2k * wh1[0][k];
            a1 += h1k * wi1[1][k] + h2k * wh1[1][k];
            a2 += h1k * wi1[2][k] + h2k * wh1[2][k];
            a3 += h1k * wi1[3][k] + h2k * wh1[3][k];
        }
        const float i1 = fast_sigmoid(a0);
        const float f1 = fast_sigmoid(a1);
        const float g1 = fast_tanh(a2);
        const float o1 = fast_sigmoid(a3);
        c2 = f1 * c2 + i1 * g1;
        h2 = o1 * fast_tanh(c2);

        // ----- regression head: reduce over the 8-lane group -----
        float p0 = h2 * wr0;
        float p1 = h2 * wr1;
        #pragma unroll
        for (int off = 1; off < 8; off <<= 1) {
            p0 += __shfl_xor(p0, off, 32);
            p1 += __shfl_xor(p1, off, 32);
        }
        if (j == 0) {
            float2 r;
            r.x = p0 + br0;
            r.y = p1 + br1;
            *(float2*)(out + ((size_t)s * BATCH + b) * 2) = r;
        }
        gin = gnext;
    }
}

extern "C" void kernel_launch(void* const* d_in, const int* in_sizes, int n_in,
                              void* d_out, int out_size, void* d_ws, size_t ws_size,
                              hipStream_t stream) {
    const float* x     = (const float*)d_in[0];
    const float* w_ih0 = (const float*)d_in[1];
    const float* w_hh0 = (const float*)d_in[2];
    const float* b_ih0 = (const float*)d_in[3];
    const float* b_hh0 = (const float*)d_in[4];
    const float* w_ih1 = (const float*)d_in[5];
    const float* w_hh1 = (const float*)d_in[6];
    const float* b_ih1 = (const float*)d_in[7];
    const float* b_hh1 = (const float*)d_in[8];
    const float* w_reg = (const float*)d_in[9];
    const float* b_reg = (const float*)d_in[10];
    float* out = (float*)d_out;
    float* xg  = (float*)d_ws;   // 512*256*32 floats = 16.8 MB

    // 131072 rows / 32 rows-per-wave = 4096 waves; 8 waves per 256-thread block.
    xgemm_wmma_kernel<<<512, 256, 0, stream>>>(x, w_ih0, b_ih0, b_hh0, xg);
    // 256 batches * 8 lanes = 2048 threads; 32 blocks of 64 for WGP spread.
    lstm_scan_kernel<<<32, 64, 0, stream>>>(xg, w_hh0, w_ih1, w_hh1,
                                            b_ih1, b_hh1, w_reg, b_reg, out);
}